// Site_MPN_83992380440679
// MI455X (gfx1250) — compile-verified
//
#include <hip/hip_runtime.h>

typedef float v2f __attribute__((ext_vector_type(2)));
typedef float v8f __attribute__((ext_vector_type(8)));
typedef unsigned int v4u __attribute__((ext_vector_type(4)));
typedef int v4i __attribute__((ext_vector_type(4)));
typedef int v8i __attribute__((ext_vector_type(8)));

#define HDIM   128
#define LSEQ   1024
#define BATCH  64
#define NNODES (BATCH * LSEQ)      // 65536
#define NEDGES (4 * NNODES)        // 262144
#define KW     12
#define KDEPTH (HDIM * KW)         // 1536
#define L1OUT  1013
#define P1OUT  337
#define L2OUT  326
#define P2OUT  81
#define NEG_BIG (-3.402823466e38f)

static __device__ __forceinline__ v8f wmma_f32x4(v2f a, v2f b, v8f c) {
  // V_WMMA_F32_16X16X4_F32 : D(16x16 f32) = A(16x4) * B(4x16) + C
  return __builtin_amdgcn_wmma_f32_16x16x4_f32(false, a, false, b, (short)0, c,
                                               false, false);
}

// ---------------- graph normalization ----------------
__global__ void k_deg(const int* __restrict__ dst, const float* __restrict__ ew,
                      float* __restrict__ deg) {
  int e = blockIdx.x * blockDim.x + threadIdx.x;
  if (e < NEDGES) atomicAdd(&deg[dst[e]], ew[e]);
}

__global__ void k_norm(const int* __restrict__ src, const int* __restrict__ dst,
                       const float* __restrict__ ew, const float* __restrict__ deg,
                       float* __restrict__ nrm) {
  int e = blockIdx.x * blockDim.x + threadIdx.x;
  if (e < NEDGES) {
    float ds = deg[src[e]], dd = deg[dst[e]];
    float is = ds > 0.f ? rsqrtf(ds) : 0.f;
    float id = dd > 0.f ? rsqrtf(dd) : 0.f;
    nrm[e] = is * ew[e] * id;
  }
}

// ---------------- embedding gather ----------------
__global__ void k_embed(const int* __restrict__ x, const float* __restrict__ emb,
                        float* __restrict__ h) {
  int t = blockIdx.x * blockDim.x + threadIdx.x;   // NNODES * 32 float4 chunks
  if (t < NNODES * (HDIM / 4)) {
    int n  = t >> 5;
    int c4 = t & 31;
    const float4* er = (const float4*)(emb + (size_t)x[n] * HDIM);
    ((float4*)(h + (size_t)n * HDIM))[c4] = er[c4];
  }
}

// ---------------- dense GEMM: Y[N,128] = X[N,128] @ W[128,128] (+bias) -------
// W staged into LDS once per workgroup via the Tensor Data Mover; one wave
// computes a 16x128 output stripe with 8 f32 WMMA accumulators, B from LDS.
__global__ void k_gemm128(const float* __restrict__ X, const float* __restrict__ W,
                          const float* __restrict__ bias, float* __restrict__ Y,
                          int hasBias) {
  __shared__ float wlds[HDIM * HDIM];              // 64 KB of the 320 KB WGP LDS

  if (threadIdx.x < 32) {                          // wave 0 issues one TDM copy
    unsigned long long ga = (unsigned long long)(uintptr_t)W;
    unsigned int lds_off = (unsigned int)(uintptr_t)&wlds[0]; // low 32b = LDS byte offset
    // D# group 0: count=1 user descriptor | lds_addr | global_addr | type=2
    v4u g0;
    g0.x = 1u;
    g0.y = lds_off;
    g0.z = (unsigned int)(ga & 0xFFFFFFFFull);
    g0.w = (unsigned int)((ga >> 32) & 0x01FFFFFFull) | (2u << 30);
    // D# group 1: data_size=4B, 1-D tensor/tile of 16384 elements, stride 16384
    v8i g1;
    g1[0] = (int)(2u << 16);                       // data_size = 2 (4 bytes)
    g1[1] = (int)(16384u << 16);                   // tensor_dim0[15:0]
    g1[2] = (int)(1u << 16);                       // tensor_dim0 hi=0, tensor_dim1=1
    g1[3] = (int)(16384u << 16);                   // tile_dim0 = 16384
    g1[4] = 0;                                     // tile_dim1/2 unused
    g1[5] = 16384;                                 // tensor_dim0_stride[31:0]
    g1[6] = 0;
    g1[7] = 0;
    v4i zz4 = {0, 0, 0, 0};
    v8i zz8 = {0, 0, 0, 0, 0, 0, 0, 0};
    __builtin_amdgcn_tensor_load_to_lds(g0, g1, zz4, zz4, zz8, 0);
    __builtin_amdgcn_s_wait_tensorcnt(0);
  }
  __syncthreads();

  int gw   = (blockIdx.x * blockDim.x + threadIdx.x) >> 5;  // wave id = row tile
  int lane = threadIdx.x & 31;
  int half = lane >> 4;
  int l15  = lane & 15;
  const float* Xr = X + (size_t)gw * 16 * HDIM;

  v8f acc[8] = {};
#pragma unroll 4
  for (int k0 = 0; k0 < HDIM; k0 += 4) {
    int ka = k0 + (half << 1);               // lanes 16-31 hold K+2 per ISA layout
    v2f a;
    a.x = Xr[l15 * HDIM + ka];
    a.y = Xr[l15 * HDIM + ka + 1];
#pragma unroll
    for (int ct = 0; ct < 8; ++ct) {
      v2f b;
      b.x = wlds[ka * HDIM + ct * 16 + l15];
      b.y = wlds[(ka + 1) * HDIM + ct * 16 + l15];
      acc[ct] = wmma_f32x4(a, b, acc[ct]);
    }
  }
  int rbase = gw * 16 + (half << 3);         // C layout: VGPR v -> M=v / M=v+8
#pragma unroll
  for (int ct = 0; ct < 8; ++ct) {
    int col  = ct * 16 + l15;
    float bv = hasBias ? bias[col] : 0.f;
#pragma unroll
    for (int v = 0; v < 8; ++v)
      Y[(size_t)(rbase + v) * HDIM + col] = acc[ct][v] + bv;
  }
}

// ---------------- edge message scatter-add ----------------
__global__ void k_scatter(const int* __restrict__ src, const int* __restrict__ dst,
                          const float* __restrict__ nrm, const float* __restrict__ hw,
                          float* __restrict__ agg) {
  int t = blockIdx.x * blockDim.x + threadIdx.x;   // NEDGES * 32 chunks of 4
  if (t >= NEDGES * (HDIM / 4)) return;
  int e  = t >> 5;
  int c  = (t & 31) << 2;
  float w = nrm[e];
  const float* m = hw + (size_t)src[e] * HDIM + c;
  float* o = agg + (size_t)dst[e] * HDIM + c;
  atomicAdd(o + 0, m[0] * w);
  atomicAdd(o + 1, m[1] * w);
  atomicAdd(o + 2, m[2] * w);
  atomicAdd(o + 3, m[3] * w);
}

__global__ void k_relu(float* __restrict__ h, int n) {
  int t = blockIdx.x * blockDim.x + threadIdx.x;
  if (t < n) h[t] = fmaxf(h[t], 0.f);
}

// ---------------- conv1d as implicit GEMM (K = 128*12) ----------------
// one wave -> 16 out-channels x 16 positions tile. A = weights, B = input patch.
__global__ void k_convgemm(const float* __restrict__ in, const float* __restrict__ w,
                           const float* __restrict__ bias, float* __restrict__ out,
                           int Lout, int LTiles, int inNodeMajor, int Lin,
                           int inBatchStride) {
  int gw   = (blockIdx.x * blockDim.x + threadIdx.x) >> 5;
  int lane = threadIdx.x & 31;
  int half = lane >> 4;
  int l15  = lane & 15;

  int perB = 8 * LTiles;
  int b    = gw / perB;
  int rem  = gw % perB;
  int ot   = rem / LTiles;
  int lt   = rem % LTiles;
  int oBase = ot * 16, lBase = lt * 16;

  const float* wrow = w + (size_t)(oBase + l15) * KDEPTH;
  v8f acc = {};
  int pmax = Lin - 1;
  for (int k0 = 0; k0 < KDEPTH; k0 += 4) {
    int ka = k0 + (half << 1);
    v2f a;
    a.x = wrow[ka];
    a.y = wrow[ka + 1];

    int i0 = ka / KW, t0 = ka - i0 * KW;
    int k1 = ka + 1;
    int i1 = k1 / KW, t1 = k1 - i1 * KW;
    int p0 = lBase + l15 + t0; p0 = p0 > pmax ? pmax : p0;
    int p1 = lBase + l15 + t1; p1 = p1 > pmax ? pmax : p1;

    v2f bf;
    if (inNodeMajor) {           // in[(b*LSEQ + pos)*128 + ch]  (ARMA output)
      bf.x = in[((size_t)b * LSEQ + p0) * HDIM + i0];
      bf.y = in[((size_t)b * LSEQ + p1) * HDIM + i1];
    } else {                     // in[b*stride + ch*Lin + pos]  (pooled, NCH)
      const float* ib = in + (size_t)b * inBatchStride;
      bf.x = ib[(size_t)i0 * Lin + p0];
      bf.y = ib[(size_t)i1 * Lin + p1];
    }
    acc = wmma_f32x4(a, bf, acc);
  }

  int col = lBase + l15;
  if (col < Lout) {
#pragma unroll
    for (int v = 0; v < 8; ++v) {
      int o = oBase + v + (half << 3);
      out[(size_t)b * HDIM * Lout + (size_t)o * Lout + col] = acc[v] + bias[o];
    }
  }
}

// ---------------- fused BN(eval) + leaky-ReLU + maxpool ----------------
__global__ void k_bnpool(const float* __restrict__ x, const float* __restrict__ g,
                         const float* __restrict__ bb, const float* __restrict__ m,
                         const float* __restrict__ vv, float* __restrict__ out,
                         int Lin, int Lpool, int k, int transposeOut) {
  int t = blockIdx.x * blockDim.x + threadIdx.x;
  int total = BATCH * HDIM * Lpool;
  if (t >= total) return;
  int lp = t % Lpool;
  int r  = t / Lpool;
  int c  = r % HDIM;
  int b  = r / HDIM;

  float scale = g[c] * rsqrtf(vv[c] + 1e-5f);
  float shift = bb[c] - m[c] * scale;
  const float* xr = x + ((size_t)b * HDIM + c) * Lin + lp * k;
  float best = NEG_BIG;
  for (int j = 0; j < k; ++j) {
    float val = xr[j] * scale + shift;
    val = val < 0.f ? 0.01f * val : val;
    best = fmaxf(best, val);
  }
  if (transposeOut)
    out[(size_t)b * Lpool * HDIM + (size_t)lp * HDIM + c] = best;  // [B, Lp, H]
  else
    out[t] = best;                                                 // [B, H, Lp]
}

// ---------------- launch ----------------
extern "C" void kernel_launch(void* const* d_in, const int* in_sizes, int n_in,
                              void* d_out, int out_size, void* d_ws, size_t ws_size,
                              hipStream_t stream) {
  (void)in_sizes; (void)n_in; (void)out_size; (void)ws_size;
  const int*   x    = (const int*)d_in[0];
  const int*   eidx = (const int*)d_in[1];
  const float* ew   = (const float*)d_in[2];
  const float* emb  = (const float*)d_in[3];
  const float* wi1  = (const float*)d_in[4];
  const float* wr1  = (const float*)d_in[5];
  const float* b1   = (const float*)d_in[6];
  const float* wi2  = (const float*)d_in[7];
  const float* wr2  = (const float*)d_in[8];
  const float* b2   = (const float*)d_in[9];
  const float* c1w  = (const float*)d_in[10];
  const float* c1b  = (const float*)d_in[11];
  const float* bn1g = (const float*)d_in[12];
  const float* bn1b = (const float*)d_in[13];
  const float* bn1m = (const float*)d_in[14];
  const float* bn1v = (const float*)d_in[15];
  const float* c2w  = (const float*)d_in[16];
  const float* c2b  = (const float*)d_in[17];
  const float* bn2g = (const float*)d_in[18];
  const float* bn2b = (const float*)d_in[19];
  const float* bn2m = (const float*)d_in[20];
  const float* bn2v = (const float*)d_in[21];

  const int* srcp = eidx;
  const int* dstp = eidx + NEDGES;

  char* ws = (char*)d_ws;
  float* deg  = (float*)(ws);                                       // 256 KB
  float* nrm  = (float*)(ws + (size_t)262144);                      // 1 MB
  float* bufA = (float*)(ws + (size_t)1310720);                     // 32 MB
  float* bufB = (float*)(ws + (size_t)1310720 + 33554432);          // 32 MB
  float* bufC = (float*)(ws + (size_t)1310720 + 2 * 33554432ull);   // 32 MB

  const int TB = 256;
  int ebl  = (NEDGES + TB - 1) / TB;
  int embl = (NNODES * (HDIM / 4) + TB - 1) / TB;
  int scbl = (NEDGES * (HDIM / 4) + TB - 1) / TB;
  int rlbl = (NNODES * HDIM + TB - 1) / TB;
  int gemmBlocks = (NNODES / 16) * 32 / TB;                         // 512

  // normalization
  (void)hipMemsetAsync(deg, 0, (size_t)NNODES * 4, stream);
  k_deg <<<ebl, TB, 0, stream>>>(dstp, ew, deg);
  k_norm<<<ebl, TB, 0, stream>>>(srcp, dstp, ew, deg, nrm);

  // embedding -> bufA
  k_embed<<<embl, TB, 0, stream>>>(x, emb, bufA);

  // ARMA layer 1: bufA -> bufC
  k_gemm128<<<gemmBlocks, TB, 0, stream>>>(bufA, wi1, nullptr, bufB, 0);
  k_gemm128<<<gemmBlocks, TB, 0, stream>>>(bufA, wr1, b1, bufC, 1);
  k_scatter<<<scbl, TB, 0, stream>>>(srcp, dstp, nrm, bufB, bufC);
  k_relu<<<rlbl, TB, 0, stream>>>(bufC, NNODES * HDIM);

  // ARMA layer 2: bufC -> bufA
  k_gemm128<<<gemmBlocks, TB, 0, stream>>>(bufC, wi2, nullptr, bufB, 0);
  k_gemm128<<<gemmBlocks, TB, 0, stream>>>(bufC, wr2, b2, bufA, 1);
  k_scatter<<<scbl, TB, 0, stream>>>(srcp, dstp, nrm, bufB, bufA);
  k_relu<<<rlbl, TB, 0, stream>>>(bufA, NNODES * HDIM);

  // conv1 (implicit GEMM): bufA -> bufC  [B,128,1013]
  int LT1 = (L1OUT + 15) / 16;                                      // 64
  int cv1Blocks = BATCH * 8 * LT1 * 32 / TB;                        // 4096
  k_convgemm<<<cv1Blocks, TB, 0, stream>>>(bufA, c1w, c1b, bufC,
                                           L1OUT, LT1, 1, LSEQ, 0);
  // bn1 + leaky + pool3: bufC -> bufB  [B,128,337]
  int p1bl = (BATCH * HDIM * P1OUT + TB - 1) / TB;
  k_bnpool<<<p1bl, TB, 0, stream>>>(bufC, bn1g, bn1b, bn1m, bn1v, bufB,
                                    L1OUT, P1OUT, 3, 0);

  // conv2: bufB -> bufC  [B,128,326]
  int LT2 = (L2OUT + 15) / 16;                                      // 21
  int cv2Blocks = BATCH * 8 * LT2 * 32 / TB;                        // 1344
  k_convgemm<<<cv2Blocks, TB, 0, stream>>>(bufB, c2w, c2b, bufC,
                                           L2OUT, LT2, 0, P1OUT, HDIM * P1OUT);
  // bn2 + leaky + pool4 + transpose: bufC -> d_out  [B,81,128]
  int p2bl = (BATCH * HDIM * P2OUT + TB - 1) / TB;
  k_bnpool<<<p2bl, TB, 0, stream>>>(bufC, bn2g, bn2b, bn2m, bn2v, (float*)d_out,
                                    L2OUT, P2OUT, 4, 1);
}